// LIFNode_35983236006441
// MI455X (gfx1250) — compile-verified
//
#include <hip/hip_runtime.h>
#include <stdint.h>

// LIF forward scan, B=128 T=32 N=8192 f32 (from reference setup_inputs).
// Memory-bound: 256 MiB total traffic -> ~11.5us floor at 23.3 TB/s.
// CDNA5 path: GLOBAL_LOAD_ASYNC_TO_LDS_B128 (ASYNCcnt-tracked) 3-deep
// prefetch pipeline through a 4-slot LDS ring, ds_load_b128 readback,
// non-temporal b128 spike stores.

typedef float v4f __attribute__((ext_vector_type(4)));

constexpr int LIF_T  = 32;
constexpr int LIF_N  = 8192;
constexpr int BLOCK  = 256;   // 8 wave32
constexpr int NBUF   = 4;     // LDS ring slots
constexpr int DEPTH  = 3;     // async prefetch distance

__device__ __forceinline__ void async_load_b128(uint32_t lds_off, uint32_t g_off,
                                                uint64_t sbase) {
  // GVS form: copies 16 bytes from global base sbase plus offset g_off to LDS at lds_off; tracked by ASYNCcnt.
  asm volatile("global_load_async_to_lds_b128 %0, %1, %2"
               :: "v"(lds_off), "v"(g_off), "s"(sbase)
               : "memory");
}

__device__ __forceinline__ float lif_step(float& m, float x) {
  // mirrors: mem = mem + (x + 0 - mem)/2 ; spike = (mem-1)>0 ; mem = spike?0:mem
  m = m + (x - m) * 0.5f;
  float s = ((m - 1.0f) > 0.0f) ? 1.0f : 0.0f;
  m = (s != 0.0f) ? 0.0f : m;
  return s;
}

__global__ __launch_bounds__(BLOCK) void lif_kernel(const float* __restrict__ X,
                                                    float* __restrict__ out) {
  __shared__ v4f smem[NBUF][BLOCK];   // 16 KB ring

  const int tid = threadIdx.x;
  const int gid = blockIdx.x * BLOCK + tid;   // over B * (N/4)
  const int Nv  = LIF_N / 4;                  // 2048 (power of two -> shifts)
  const int b   = gid / Nv;
  const int nv  = gid - b * Nv;

  // element offset of (b, t=0, n=4*nv)
  const size_t   baseElem = (size_t)b * LIF_T * LIF_N + (size_t)nv * 4;
  const uint32_t gOff0    = (uint32_t)(baseElem * sizeof(float)); // < 128 MiB, fits u32
  const uint32_t strideB  = (uint32_t)(LIF_N * sizeof(float));    // 32 KiB per timestep
  const uint64_t xbase    = (uint64_t)(uintptr_t)X;

  // Prologue: launch slices 0..DEPTH-1 into the ring.
#pragma unroll
  for (int i = 0; i < DEPTH; ++i) {
    uint32_t lds = (uint32_t)(uintptr_t)&smem[i][tid]; // low 32 bits = LDS offset
    async_load_b128(lds, gOff0 + (uint32_t)i * strideB, xbase);
  }

  float m0 = 0.0f, m1 = 0.0f, m2 = 0.0f, m3 = 0.0f;
  float* op = out + baseElem;

#pragma unroll
  for (int t = 0; t < LIF_T; ++t) {
    // Async loads complete in order; with (3 + t) issued so far, ASYNCcnt<=2
    // guarantees slice t has landed. Tail iterations just drain everything.
    if (t + DEPTH < LIF_T)
      asm volatile("s_wait_asynccnt 0x2" ::: "memory");
    else
      asm volatile("s_wait_asynccnt 0x0" ::: "memory");

    v4f x = smem[t & (NBUF - 1)][tid];   // ds_load_b128, own 16B -> no barrier

    // Keep the pipe full: issue slice t+DEPTH into slot (t+DEPTH)%NBUF.
    if (t + DEPTH < LIF_T) {
      uint32_t lds = (uint32_t)(uintptr_t)&smem[(t + DEPTH) & (NBUF - 1)][tid];
      async_load_b128(lds, gOff0 + (uint32_t)(t + DEPTH) * strideB, xbase);
    }

    v4f sp;
    sp.x = lif_step(m0, x.x);
    sp.y = lif_step(m1, x.y);
    sp.z = lif_step(m2, x.z);
    sp.w = lif_step(m3, x.w);

    // Write-once stream: non-temporal b128 store.
    __builtin_nontemporal_store(sp, (v4f*)(op + (size_t)t * LIF_N));
  }
}

extern "C" void kernel_launch(void* const* d_in, const int* in_sizes, int n_in,
                              void* d_out, int out_size, void* d_ws, size_t ws_size,
                              hipStream_t stream) {
  (void)n_in; (void)d_ws; (void)ws_size; (void)out_size;
  const float* X  = (const float*)d_in[0];
  float*       O  = (float*)d_out;

  // B derived from flat size; T, N fixed per reference setup.
  const int B = in_sizes[0] / (LIF_T * LIF_N);     // = 128
  const int totalVec = B * (LIF_N / 4);            // 262144 threads
  const int blocks   = totalVec / BLOCK;           // 1024

  lif_kernel<<<blocks, BLOCK, 0, stream>>>(X, O);
}